// RGCN2_83476984365515
// MI455X (gfx1250) — compile-verified
//
#include <hip/hip_runtime.h>
#include <hip/hip_bf16.h>

typedef float v2f __attribute__((ext_vector_type(2)));
typedef float v8f __attribute__((ext_vector_type(8)));

__device__ __forceinline__ void atomicAddF32(float* p, float v) {
    (void)__hip_atomic_fetch_add(p, v, __ATOMIC_RELAXED, __HIP_MEMORY_SCOPE_AGENT);
}

// ---------------------------------------------------------------- utilities
__global__ void zero_f(float* __restrict__ p, long n) {
    long i = (long)blockIdx.x * blockDim.x + threadIdx.x;
    if (i < n) p[i] = 0.0f;
}

__global__ void count_deg(const int* __restrict__ src, const int* __restrict__ dst,
                          float* __restrict__ dout, float* __restrict__ din, int E) {
    int e = blockIdx.x * blockDim.x + threadIdx.x;
    if (e < E) {
        atomicAddF32(dout + src[e], 1.0f);
        atomicAddF32(din + dst[e], 1.0f);
    }
}

__global__ void rsqrt_deg(float* __restrict__ p, long n) {
    long i = (long)blockIdx.x * blockDim.x + threadIdx.x;
    if (i < n) p[i] = rsqrtf(fmaxf(p[i], 1.0f));
}

// --------------------------------------------------- edge scatter (gather+atomic)
// one edge per wave (32 threads, uniform src/dst); each lane moves 4 consecutive
// floats of the 128-wide row via b128 gather + 4 f32 hardware atomics.
// Folds the rsqrt(out-degree) scaling of the source row into the gather.
__global__ void scatter_edges(const float* __restrict__ x, const int* __restrict__ src,
                              const int* __restrict__ dst, const float* __restrict__ rs_out,
                              float* __restrict__ agg, int E) {
    long tid = (long)blockIdx.x * blockDim.x + threadIdx.x;
    int e = (int)(tid >> 5);
    int q = (int)(tid & 31);
    if (e >= E) return;
    int s = src[e];
    int d = dst[e];
    float sc = rs_out[s];
    const float4 v = *(const float4*)(x + (size_t)s * 128 + q * 4);
    float* b = agg + (size_t)d * 128 + q * 4;
    atomicAddF32(b + 0, v.x * sc);
    atomicAddF32(b + 1, v.y * sc);
    atomicAddF32(b + 2, v.z * sc);
    atomicAddF32(b + 3, v.w * sc);
}

// --------------------------------------------------- WMMA f32 GEMM: C = (A*rowScale) @ W + bias
// A: M x K (lda), W: K x 128 row-major, C: M x 128 written at C[r*ldc + colOff + c].
// 256 threads = 8 waves; each wave computes TWO 16x128 strips (block covers 256 rows)
// so every B fragment (one aligned ds_load_b64 from a K-pair-swizzled LDS tile of W)
// feeds two V_WMMA_F32_16X16X4_F32 ops.
__global__ __launch_bounds__(256) void gemm_wmma_n128(
        const float* __restrict__ A, int lda,
        const float* __restrict__ rowScale,   // nullable: rsqrt(in-degree) per row
        const float* __restrict__ W,
        const float* __restrict__ bias,       // nullable, length 128
        float* __restrict__ C, int ldc, int colOff,
        int M, int K, int doRelu)
{
    // swizzled W tile: sW[(k>>1)*256 + 2*col + (k&1)], k in [0,64), col in [0,128)
    __shared__ float sW[64 * 128];            // 32 KB
    const int wave = threadIdx.x >> 5;
    const int lane = threadIdx.x & 31;
    const int half = lane >> 4;               // 0: K {0,1} / M 0..7 ; 1: K {2,3} / M 8..15
    const int l    = lane & 15;
    const int rowBase0 = blockIdx.x * 256 + wave * 16;
    const int rowBase1 = rowBase0 + 128;

    int ar0 = rowBase0 + l; if (ar0 > M - 1) ar0 = M - 1;   // clamp; masked at store
    int ar1 = rowBase1 + l; if (ar1 > M - 1) ar1 = M - 1;
    const float rs0 = rowScale ? rowScale[ar0] : 1.0f;
    const float rs1 = rowScale ? rowScale[ar1] : 1.0f;
    const float* A0 = A + (size_t)ar0 * lda;
    const float* A1 = A + (size_t)ar1 * lda;

    v8f acc0[8] = {};
    v8f acc1[8] = {};

    for (int kk = 0; kk < K; kk += 64) {
        __syncthreads();
        // stage 64x128 W tile swizzled into K-pairs; unit u: p = row pair, cp = col pair
        for (int u = threadIdx.x; u < 2048; u += 256) {
            int p = u >> 6, cp = u & 63;
            const float* w0 = W + (size_t)(kk + 2 * p) * 128 + 2 * cp;
            float2 lo = *(const float2*)(w0);
            float2 hi = *(const float2*)(w0 + 128);
            float4 t; t.x = lo.x; t.y = hi.x; t.z = lo.y; t.w = hi.y;
            *(float4*)&sW[p * 256 + 4 * cp] = t;
        }
        __syncthreads();

        for (int k0 = 0; k0 < 64; k0 += 4) {
            const int ka = k0 + 2 * half;
            float2 av0 = *(const float2*)(A0 + kk + ka);
            float2 av1 = *(const float2*)(A1 + kk + ka);
            v2f a0; a0.x = av0.x * rs0; a0.y = av0.y * rs0;
            v2f a1; a1.x = av1.x * rs1; a1.y = av1.y * rs1;
            const float* sB = sW + (size_t)(ka >> 1) * 256;
#pragma unroll
            for (int t = 0; t < 8; ++t) {
                const int col = t * 16 + l;
                v2f b = *(const v2f*)(sB + 2 * col);   // one aligned ds_load_b64
                acc0[t] = __builtin_amdgcn_wmma_f32_16x16x4_f32(
                    false, a0, false, b, (short)0, acc0[t], false, false);
                acc1[t] = __builtin_amdgcn_wmma_f32_16x16x4_f32(
                    false, a1, false, b, (short)0, acc1[t], false, false);
            }
        }
    }

    // epilogue: C/D layout — VGPR i: row = tileBase + half*8 + i, col = l + 16*t
#pragma unroll
    for (int t = 0; t < 8; ++t) {
        const int col = t * 16 + l;
        const float bv = bias ? bias[col] : 0.0f;
#pragma unroll
        for (int i = 0; i < 8; ++i) {
            int r0 = rowBase0 + half * 8 + i;
            if (r0 < M) {
                float v = acc0[t][i] + bv;
                if (doRelu) v = fmaxf(v, 0.0f);
                C[(size_t)r0 * ldc + colOff + col] = v;
            }
            int r1 = rowBase1 + half * 8 + i;
            if (r1 < M) {
                float v = acc1[t][i] + bv;
                if (doRelu) v = fmaxf(v, 0.0f);
                C[(size_t)r1 * ldc + colOff + col] = v;
            }
        }
    }
}

// --------------------------------------------------- BatchNorm stats + head
__global__ void bn_stats(const float* __restrict__ h, float* __restrict__ colsum,
                         float* __restrict__ colsq, int M) {
    int c = threadIdx.x;                      // 128 threads = 128 columns
    float s = 0.0f, q = 0.0f;
    for (int r = blockIdx.x; r < M; r += gridDim.x) {
        float v = h[(size_t)r * 128 + c];
        s += v;
        q += v * v;
    }
    atomicAddF32(colsum + c, s);
    atomicAddF32(colsq + c, q);
}

__global__ void bn_finalize(const float* __restrict__ sum, const float* __restrict__ sq,
                            const float* __restrict__ gamma, const float* __restrict__ beta,
                            float* __restrict__ scale, float* __restrict__ shift, int M) {
    int c = threadIdx.x;
    float invM = 1.0f / (float)M;
    float mu  = sum[c] * invM;
    float var = sq[c] * invM - mu * mu;       // biased variance, as in torch BN
    float sc  = gamma[c] * rsqrtf(var + 1e-5f);
    scale[c] = sc;
    shift[c] = beta[c] - mu * sc;
}

// out[row,{0,1}] = sum_k relu(h1[row,k]*scale[k]+shift[k]) * Wm2[k,{0,1}]
__global__ void mlp_out_kernel(const float* __restrict__ h1, const float* __restrict__ scale,
                               const float* __restrict__ shift, const float* __restrict__ Wm2,
                               float* __restrict__ out, int M) {
    int wave = threadIdx.x >> 5;
    int lane = threadIdx.x & 31;
    int row = blockIdx.x * 8 + wave;
    if (row >= M) return;
    const float* hr = h1 + (size_t)row * 128;
    float a0 = 0.0f, a1 = 0.0f;
#pragma unroll
    for (int k = lane; k < 128; k += 32) {
        float v = fmaxf(hr[k] * scale[k] + shift[k], 0.0f);
        a0 += v * Wm2[2 * k];
        a1 += v * Wm2[2 * k + 1];
    }
    for (int off = 16; off > 0; off >>= 1) {
        a0 += __shfl_down(a0, off, 32);
        a1 += __shfl_down(a1, off, 32);
    }
    if (lane == 0) {
        out[(size_t)row * 2 + 0] = a0;
        out[(size_t)row * 2 + 1] = a1;
    }
}

// ---------------------------------------------------------------- launch
extern "C" void kernel_launch(void* const* d_in, const int* in_sizes, int n_in,
                              void* d_out, int out_size, void* d_ws, size_t ws_size,
                              hipStream_t stream) {
    const float* xA    = (const float*)d_in[0];
    const float* xB    = (const float*)d_in[1];
    const int*   src_r0 = (const int*)d_in[2];
    const int*   dst_r0 = (const int*)d_in[3];
    const int*   src_r1 = (const int*)d_in[4];
    const int*   dst_r1 = (const int*)d_in[5];
    const float* W1_r0 = (const float*)d_in[6];
    const float* b1_r0 = (const float*)d_in[7];
    const float* W1_r1 = (const float*)d_in[8];
    const float* b1_r1 = (const float*)d_in[9];
    const float* W2_r0 = (const float*)d_in[10];
    const float* b2_r0 = (const float*)d_in[11];
    const float* W2_r1 = (const float*)d_in[12];
    const float* b2_r1 = (const float*)d_in[13];
    const float* Wm1   = (const float*)d_in[14];
    const float* gamma = (const float*)d_in[15];
    const float* beta  = (const float*)d_in[16];
    const float* Wm2   = (const float*)d_in[17];
    float* out = (float*)d_out;

    const int N = in_sizes[0] / 128;
    const int E = in_sizes[2];

    // workspace layout (floats)
    float* ws   = (float*)d_ws;
    float* rs   = ws;                              // 4N: [dout_r0 | din_r0 | dout_r1 | din_r1]
    float* hA   = rs + (size_t)4 * N;              // N*128
    float* hB   = hA + (size_t)N * 128;            // N*128
    float* Hcat = hB + (size_t)N * 128;            // N*256 : [h2A | h2B]
    float* agg  = Hcat + (size_t)N * 256;          // N*128 scratch, reused as h1
    float* stats = agg + (size_t)N * 128;          // 512: sum|sumsq|scale|shift

    auto cdiv = [](long a, long b) { return (unsigned)((a + b - 1) / b); };
    dim3 blk(256);
    const long n4   = (long)4 * N;
    const long nAgg = (long)N * 128;
    const unsigned gS = cdiv((long)E * 32, 256);
    const unsigned gG = cdiv(N, 256);

    // degrees -> rsqrt(max(deg,1))
    zero_f<<<cdiv(n4, 256), blk, 0, stream>>>(rs, n4);
    count_deg<<<cdiv(E, 256), blk, 0, stream>>>(src_r0, dst_r0, rs, rs + N, E);
    count_deg<<<cdiv(E, 256), blk, 0, stream>>>(src_r1, dst_r1, rs + 2L * N, rs + 3L * N, E);
    rsqrt_deg<<<cdiv(n4, 256), blk, 0, stream>>>(rs, n4);

    // conv1, r0: A->B  => hB = relu(norm-agg(xA) @ W1_r0 + b1_r0)
    zero_f<<<cdiv(nAgg, 256), blk, 0, stream>>>(agg, nAgg);
    scatter_edges<<<gS, blk, 0, stream>>>(xA, src_r0, dst_r0, rs, agg, E);
    gemm_wmma_n128<<<gG, blk, 0, stream>>>(agg, 128, rs + N, W1_r0, b1_r0, hB, 128, 0, N, 128, 1);

    // conv1, r1: B->A  => hA
    zero_f<<<cdiv(nAgg, 256), blk, 0, stream>>>(agg, nAgg);
    scatter_edges<<<gS, blk, 0, stream>>>(xB, src_r1, dst_r1, rs + 2L * N, agg, E);
    gemm_wmma_n128<<<gG, blk, 0, stream>>>(agg, 128, rs + 3L * N, W1_r1, b1_r1, hA, 128, 0, N, 128, 1);

    // conv2, r0: hA->B => h2B into Hcat cols [128,256)
    zero_f<<<cdiv(nAgg, 256), blk, 0, stream>>>(agg, nAgg);
    scatter_edges<<<gS, blk, 0, stream>>>(hA, src_r0, dst_r0, rs, agg, E);
    gemm_wmma_n128<<<gG, blk, 0, stream>>>(agg, 128, rs + N, W2_r0, b2_r0, Hcat, 256, 128, N, 128, 0);

    // conv2, r1: hB->A => h2A into Hcat cols [0,128)
    zero_f<<<cdiv(nAgg, 256), blk, 0, stream>>>(agg, nAgg);
    scatter_edges<<<gS, blk, 0, stream>>>(hB, src_r1, dst_r1, rs + 2L * N, agg, E);
    gemm_wmma_n128<<<gG, blk, 0, stream>>>(agg, 128, rs + 3L * N, W2_r1, b2_r1, Hcat, 256, 0, N, 128, 0);

    // MLP layer 1: h1 = Hcat(N,256) @ Wm1(256,128)   (h1 reuses agg)
    gemm_wmma_n128<<<gG, blk, 0, stream>>>(Hcat, 256, nullptr, Wm1, nullptr, agg, 128, 0, N, 256, 0);

    // BatchNorm stats + head
    zero_f<<<1, blk, 0, stream>>>(stats, 256);
    bn_stats<<<128, 128, 0, stream>>>(agg, stats, stats + 128, N);
    bn_finalize<<<1, 128, 0, stream>>>(stats, stats + 128, gamma, beta, stats + 256, stats + 384, N);
    mlp_out_kernel<<<cdiv(N, 8), blk, 0, stream>>>(agg, stats + 256, stats + 384, Wm2, out, N);
}